// PaGNNConv_78606491452014
// MI455X (gfx1250) — compile-verified
//
#include <hip/hip_runtime.h>

typedef __attribute__((ext_vector_type(2))) float v2f;
typedef __attribute__((ext_vector_type(8))) float v8f;

// ---------------------------------------------------------------------------
// Workspace layout (floats):
//   [0, N)          deg  -> overwritten in place with dinv
//   [N, 2N)         rowsum
//   [2N, 2N+64N)    agg   (A @ (mask*x))
//   [2N+64N, ...)   denom (A @ mask)
// ---------------------------------------------------------------------------

__global__ void zero_kernel(float* __restrict__ p, long n) {
  long i = (long)blockIdx.x * blockDim.x + threadIdx.x;
  long stride = (long)gridDim.x * blockDim.x;
  for (; i < n; i += stride) p[i] = 0.0f;
}

__global__ void degree_kernel(const int* __restrict__ ei, float* __restrict__ deg, int E) {
  int e = blockIdx.x * blockDim.x + threadIdx.x;
  if (e >= E) return;
  atomicAdd(&deg[ei[E + e]], 1.0f);  // degree over col, counts are exact in f32
}

__global__ void dinv_kernel(float* __restrict__ deg, int n) {
  int i = blockIdx.x * blockDim.x + threadIdx.x;
  if (i >= n) return;
  float d = deg[i];
  deg[i] = (d > 0.0f) ? rsqrtf(d) : 0.0f;
}

// One wave (32 lanes) per edge; lanes cover features {l, l+32} -> coalesced atomics.
__global__ void scatter_kernel(const float* __restrict__ x, const float* __restrict__ mask,
                               const int* __restrict__ ei, const float* __restrict__ dinv,
                               float* __restrict__ rowsum, float* __restrict__ agg,
                               float* __restrict__ denom, int E) {
  int t = blockIdx.x * blockDim.x + threadIdx.x;
  int e = t >> 5;
  if (e >= E) return;
  int lane = t & 31;
  int r = ei[e];
  int c = ei[E + e];
  float w = dinv[r] * dinv[c];
  if (w == 0.0f) return;  // wave-uniform: adding zero is a no-op, skip atomics
  if (lane == 0) atomicAdd(&rowsum[r], w);
  const float* xr = x + (long)c * 64;
  const float* mr = mask + (long)c * 64;
  float* ar = agg + (long)r * 64;
  float* dr = denom + (long)r * 64;
#pragma unroll
  for (int h = 0; h < 2; ++h) {
    int d = lane + 32 * h;
    float mv = mr[d];
    if (mv != 0.0f) {  // mask is 0/1, ~30% zeros: skip zero-contribution atomics
      float xv = xr[d];
      xv = (xv == xv) ? xv : 0.0f;  // nan_to_num on x
      atomicAdd(&ar[d], w * mv * xv);
      atomicAdd(&dr[d], w * mv);
    }
  }
}

// Branch-free guarded divide: rcp + one Newton refinement (~1 ulp), lowers to
// v_rcp_f32 + fma + v_cndmask instead of the IEEE div sequence w/ EXEC branches.
__device__ __forceinline__ float guarded_div(float num, float den) {
  float r = __builtin_amdgcn_rcpf(den);
  float q = num * r;
  q = __builtin_fmaf(__builtin_fmaf(-q, den, num), r, q);  // one NR step
  return (den > 0.0f) ? q : 0.0f;
}

// One wave per 16-node tile: ratio[16,64] (computed on the fly) @ W^T + b via
// V_WMMA_F32_16X16X4_F32, 16 k-steps x 4 output tiles, bias pre-loaded into C.
__global__ void __launch_bounds__(128) finalize_kernel(
    const float* __restrict__ rowsum, const float* __restrict__ agg,
    const float* __restrict__ denom, const float* __restrict__ Wm,
    const float* __restrict__ bias, float* __restrict__ out, int tiles) {
  int wave = (blockIdx.x * blockDim.x + threadIdx.x) >> 5;
  if (wave >= tiles) return;  // whole-wave exit: EXEC stays all-1s for WMMA
  int lane = threadIdx.x & 31;
  int m  = lane & 15;   // A: row M ; B: col o%16 ; D: col N
  int kh = lane >> 4;   // K half-select within each 4-chunk
  long base = (long)wave * 16;

  float rs = rowsum[base + m];

  v8f acc[4];
#pragma unroll
  for (int nt = 0; nt < 4; ++nt) {
    float bv = bias[nt * 16 + m];
#pragma unroll
    for (int v = 0; v < 8; ++v) acc[nt][v] = bv;  // D = ratio@W^T + b
  }

  const float* aggRow = agg + (base + m) * 64;
  const float* denRow = denom + (base + m) * 64;

#pragma unroll
  for (int kc = 0; kc < 16; ++kc) {
    int d0 = 4 * kc + 2 * kh;  // even -> 8B-aligned float2 loads
    v2f av = *(const v2f*)(aggRow + d0);
    v2f dv = *(const v2f*)(denRow + d0);
    v2f a;
    a.x = guarded_div(rs * av.x, dv.x);
    a.y = guarded_div(rs * av.y, dv.y);
#pragma unroll
    for (int nt = 0; nt < 4; ++nt) {
      v2f bb = *(const v2f*)(Wm + (long)(nt * 16 + m) * 64 + d0);  // B[k][o] = W[o][k]
      acc[nt] = __builtin_amdgcn_wmma_f32_16x16x4_f32(
          /*neg_a=*/false, a, /*neg_b=*/false, bb,
          /*c_mod=*/(short)0, acc[nt], /*reuse_a=*/false, /*reuse_b=*/false);
    }
  }

#pragma unroll
  for (int v = 0; v < 8; ++v) {
    long row = base + v + 8 * kh;  // C/D layout: VGPR v <-> M = v + 8*kh
#pragma unroll
    for (int nt = 0; nt < 4; ++nt)
      out[row * 64 + nt * 16 + m] = acc[nt][v];
  }
}

// Scalar fallback for a partial tile (not hit for N=50000, kept for safety).
__global__ void finalize_scalar(const float* __restrict__ rowsum, const float* __restrict__ agg,
                                const float* __restrict__ denom, const float* __restrict__ Wm,
                                const float* __restrict__ bias, float* __restrict__ out,
                                int startNode, int N) {
  int idx = blockIdx.x * blockDim.x + threadIdx.x;
  int node = startNode + idx / 64;
  int o = idx % 64;
  if (node >= N) return;
  float rs = rowsum[node];
  float s = bias[o];
  for (int d = 0; d < 64; ++d) {
    float den = denom[(long)node * 64 + d];
    float r = (den > 0.0f) ? rs * agg[(long)node * 64 + d] / den : 0.0f;
    s += r * Wm[o * 64 + d];
  }
  out[(long)node * 64 + o] = s;
}

extern "C" void kernel_launch(void* const* d_in, const int* in_sizes, int n_in,
                              void* d_out, int out_size, void* d_ws, size_t ws_size,
                              hipStream_t stream) {
  const float* x    = (const float*)d_in[0];
  const float* mask = (const float*)d_in[1];
  const float* Wm   = (const float*)d_in[2];
  const float* bias = (const float*)d_in[3];
  const int*   ei   = (const int*)d_in[4];
  float* out = (float*)d_out;

  const int N = in_sizes[0] / 64;
  const int E = in_sizes[4] / 2;

  float* ws     = (float*)d_ws;
  float* deg    = ws;               // N (becomes dinv in place)
  float* rowsum = ws + N;           // N
  float* agg    = ws + 2L * N;      // 64N
  float* denom  = agg + 64L * N;    // 64N

  long zn = 2L * N + 128L * N;
  zero_kernel<<<2048, 256, 0, stream>>>(ws, zn);
  degree_kernel<<<(E + 255) / 256, 256, 0, stream>>>(ei, deg, E);
  dinv_kernel<<<(N + 255) / 256, 256, 0, stream>>>(deg, N);

  long st = (long)E * 32;  // one wave per edge
  scatter_kernel<<<(int)((st + 255) / 256), 256, 0, stream>>>(x, mask, ei, deg,
                                                              rowsum, agg, denom, E);

  int tiles = N / 16;
  if (tiles > 0)
    finalize_kernel<<<(tiles + 3) / 4, 128, 0, stream>>>(rowsum, agg, denom, Wm, bias, out, tiles);
  int rem = N - tiles * 16;
  if (rem > 0)
    finalize_scalar<<<(rem * 64 + 255) / 256, 256, 0, stream>>>(rowsum, agg, denom, Wm, bias,
                                                                out, tiles * 16, N);
}